// RejectionSamplerBaseline_22445499089620
// MI455X (gfx1250) — compile-verified
//
#include <hip/hip_runtime.h>
#include <hip/hip_bf16.h>
#include <stdint.h>

#define EPSF 1e-10f
#define TILE 256      // elements per search tile (one wave: 32 lanes x 8 floats)
#define MAXNT 512     // max tiles per row supported (V <= 131072)

// ---- CDNA5 async global->LDS DMA (gfx1250) -------------------------------
// Per-lane: copies 16 bytes from the global address in vaddr to the LDS address in vdst, tracked by ASYNCcnt.
__device__ __forceinline__ unsigned lds_off_of(const void* p) {
  // Flat pointers to LDS carry the byte offset in addr[31:0] (ISA 10.2).
  return (unsigned)(unsigned long long)(uintptr_t)p;
}
__device__ __forceinline__ void async_ld_b128(unsigned lds_byte_off, const void* g) {
  asm volatile("global_load_async_to_lds_b128 %0, %1, off"
               :: "v"(lds_byte_off), "v"(g) : "memory");
}
__device__ __forceinline__ void wait_async_le0() {
  asm volatile("s_wait_asynccnt 0" ::: "memory");
}
__device__ __forceinline__ void wait_async_le4() {
  // 4 newest ops (next tile, still in flight) may remain outstanding.
  asm volatile("s_wait_asynccnt 4" ::: "memory");
}

// ---- Kernel 1: one block (8 waves) per row n -----------------------------
__global__ __launch_bounds__(256)
void rs_row_kernel(const int* __restrict__ tok,
                   const float* __restrict__ dp,   // draft_probs  [N,V]
                   const float* __restrict__ tp,   // target_probs [N,V]
                   const float* __restrict__ us,   // uniform_samples [N]
                   const float* __restrict__ ur,   // uniform_recover [N]
                   int* __restrict__ wsAccept,     // [N]
                   int* __restrict__ wsRec,        // [N]
                   int V)
{
  const int row  = blockIdx.x;
  const int tid  = threadIdx.x;
  const int wave = tid >> 5;
  const int lane = tid & 31;
  const int NT   = (V + TILE - 1) / TILE;

  __shared__ float sAdj[MAXNT];                    // per-tile sum of max(t-d,0)
  __shared__ float sT[MAXNT];                      // per-tile sum of t
  __shared__ float sWaveTot[8];                    // per-wave partial of total s
  __shared__ __align__(16) float stage[8][2][2 * TILE]; // [wave][buf][tgt|dft]
  __shared__ int   shTi, shMode;
  __shared__ float shBase, shThresh;

  const float* gT = tp + (size_t)row * V;
  const float* gD = dp + (size_t)row * V;

  // Acceptance test (cheap gather)
  if (tid == 0) {
    int t = tok[row];
    float r = gT[t] / fmaxf(gD[t], EPSF);
    wsAccept[row] = (us[row] < fminf(1.0f, r)) ? 1 : 0;
  }

  // ---- Phase 1: per-tile partial sums, async DMA double-buffered --------
  const unsigned sbase0 = lds_off_of(&stage[wave][0][0]);
  const unsigned loff   = (unsigned)lane * 16u;

  // prologue: issue first tile's DMA (if it is a full tile)
  {
    const int t0 = wave;
    if (t0 < NT && t0 * TILE + TILE <= V) {
      const int e0 = t0 * TILE;
      async_ld_b128(sbase0 +         loff, gT + e0 +       lane * 4);
      async_ld_b128(sbase0 +  512u + loff, gT + e0 + 128 + lane * 4);
      async_ld_b128(sbase0 + 1024u + loff, gD + e0 +       lane * 4);
      async_ld_b128(sbase0 + 1536u + loff, gD + e0 + 128 + lane * 4);
    }
  }

  float waveTot = 0.f;
  for (int t = wave, it = 0; t < NT; t += 8, ++it) {
    const int  e0       = t * TILE;
    const int  p        = it & 1;
    const int  tn       = t + 8;
    const bool thisFull = (e0 + TILE <= V);
    const bool nextFull = (tn < NT) && (tn * TILE + TILE <= V);

    // issue next tile into the other buffer before consuming this one
    if (nextFull) {
      const unsigned nb = sbase0 + (unsigned)(p ^ 1) * (2 * TILE * 4);
      const int      en = tn * TILE;
      async_ld_b128(nb +         loff, gT + en +       lane * 4);
      async_ld_b128(nb +  512u + loff, gT + en + 128 + lane * 4);
      async_ld_b128(nb + 1024u + loff, gD + en +       lane * 4);
      async_ld_b128(nb + 1536u + loff, gD + en + 128 + lane * 4);
    }

    float aSum = 0.f, tSum = 0.f;
    if (thisFull) {
      if (nextFull) wait_async_le4(); else wait_async_le0();
      const float4* st4 = reinterpret_cast<const float4*>(&stage[wave][p][0])    + lane * 2;
      const float4* sd4 = reinterpret_cast<const float4*>(&stage[wave][p][TILE]) + lane * 2;
#pragma unroll
      for (int k = 0; k < 2; ++k) {
        float4 tv = st4[k];
        float4 dv = sd4[k];
        tSum += tv.x + tv.y + tv.z + tv.w;
        aSum += fmaxf(tv.x - dv.x, 0.f) + fmaxf(tv.y - dv.y, 0.f)
              + fmaxf(tv.z - dv.z, 0.f) + fmaxf(tv.w - dv.w, 0.f);
      }
    } else {                                         // ragged tail tile
      for (int j = 0; j < 8; ++j) {
        int idx = e0 + lane * 8 + j;
        if (idx < V) {
          float tv = gT[idx], dv = gD[idx];
          tSum += tv;
          aSum += fmaxf(tv - dv, 0.f);
        }
      }
    }
#pragma unroll
    for (int m = 16; m > 0; m >>= 1) {
      aSum += __shfl_xor(aSum, m, 32);
      tSum += __shfl_xor(tSum, m, 32);
    }
    if (lane == 0) { sAdj[t] = aSum; sT[t] = tSum; }
    waveTot += aSum;                                 // all lanes hold full tile sum
  }
  if (lane == 0) sWaveTot[wave] = waveTot;
  __syncthreads();

  // ---- Phase 2a: find the CDF crossing tile (serial, early-exit) --------
  if (tid == 0) {
    float s = 0.f;
#pragma unroll
    for (int w = 0; w < 8; ++w) s += sWaveTot[w];
    const int   mode = (s > EPSF) ? 1 : 0;           // 1: residual dist, 0: target fallback
    const float u    = ur[row];
    const float thr  = mode ? u * s : u;             // unnormalized threshold
    float run = 0.f, base = 0.f;
    int ti = -1;
    for (int t = 0; t < NT; ++t) {
      float v = mode ? sAdj[t] : sT[t];
      if (run + v >= thr) { ti = t; base = run; break; }
      run += v;
    }
    shTi = ti; shMode = mode; shBase = base; shThresh = thr;
  }
  __syncthreads();

  // ---- Phase 2b: wave 0 counts inside the crossing tile (1KB, L2-hot) ----
  if (wave == 0) {
    const int ti = shTi;
    int count;
    if (ti < 0) {
      count = V;                                     // threshold above total mass
    } else {
      const int   mode = shMode;
      const float thr  = shThresh;
      const int   e0   = ti * TILE;
      float vals[8];
      float ls = 0.f;
#pragma unroll
      for (int j = 0; j < 8; ++j) {
        int idx = e0 + lane * 8 + j;
        float v = 0.f;
        if (idx < V) {
          float tv = gT[idx];
          v = mode ? fmaxf(tv - gD[idx], 0.f) : tv;
        }
        vals[j] = v;
        ls += v;
      }
      // wave inclusive scan of lane sums -> exclusive lane prefix
      float inc = ls;
#pragma unroll
      for (int d = 1; d < 32; d <<= 1) {
        float y = __shfl_up(inc, d, 32);
        if (lane >= d) inc += y;
      }
      float run = shBase + (inc - ls);
      int c = 0;
#pragma unroll
      for (int j = 0; j < 8; ++j) {
        run += vals[j];
        int idx = e0 + lane * 8 + j;
        if (idx < V && run < thr) c++;
      }
#pragma unroll
      for (int m = 16; m > 0; m >>= 1) c += __shfl_xor(c, m, 32);
      count = e0 + c;                                // all earlier tiles fully below thr
    }
    if (lane == 0) wsRec[row] = min(count, V - 1);
  }
}

// ---- Kernel 2: per-sequence finalize (B=128 lanes) -----------------------
__global__ void rs_finalize_kernel(const int* __restrict__ tok,
                                   const int* __restrict__ bonus,
                                   const int* __restrict__ wsAccept,
                                   const int* __restrict__ wsRec,
                                   float* __restrict__ out,  // concatenated outputs
                                   int B, int K)
{
  int b = blockIdx.x * blockDim.x + threadIdx.x;
  if (b >= B) return;
  int nacc = 0;
  bool run = true;
  for (int k = 0; k < K; ++k) {
    if (run && wsAccept[b * K + k]) nacc++;
    else run = false;
  }
  const bool all  = (nacc == K);
  const int recat = wsRec[b * K + min(nacc, K - 1)];
  const int ftok  = all ? bonus[b] : recat;

  float* orow = out + (size_t)b * (K + 1);
  for (int j = 0; j <= K; ++j) {
    int v;
    if (j < nacc)       v = tok[b * K + j];
    else if (j == nacc) v = ftok;
    else                v = -1;                      // PLACEHOLDER_TOKEN_ID
    orow[j] = (float)v;
  }
  float* tail = out + (size_t)B * (K + 1);
  tail[b]         = (float)(nacc + 1);               // num_accepted
  tail[B + b]     = (float)nacc;                     // accepted_counts
  tail[2 * B + b] = all ? 0.f : 1.f;                 // recovered_counts
  tail[3 * B + b] = all ? 1.f : 0.f;                 // bonus_counts
}

extern "C" void kernel_launch(void* const* d_in, const int* in_sizes, int n_in,
                              void* d_out, int out_size, void* d_ws, size_t ws_size,
                              hipStream_t stream) {
  const int*   tok   = (const int*)d_in[0];
  // d_in[1] = num_draft_tokens (scalar) — derived below instead
  const float* dp    = (const float*)d_in[2];
  const float* tp    = (const float*)d_in[3];
  const int*   bonus = (const int*)d_in[4];
  const float* us    = (const float*)d_in[5];
  const float* ur    = (const float*)d_in[6];

  const int N = in_sizes[0];
  const int V = in_sizes[2] / N;
  const int B = in_sizes[4];
  const int K = N / B;

  int* wsAccept = (int*)d_ws;        // [N]
  int* wsRec    = wsAccept + N;      // [N]

  rs_row_kernel<<<N, 256, 0, stream>>>(tok, dp, tp, us, ur, wsAccept, wsRec, V);
  rs_finalize_kernel<<<(B + 127) / 128, 128, 0, stream>>>(tok, bonus, wsAccept,
                                                          wsRec, (float*)d_out, B, K);
}